// TrustAwareTemporalGNN_76836964926269
// MI455X (gfx1250) — compile-verified
//
#include <hip/hip_runtime.h>
#include <math.h>

// ---------------------------------------------------------------------------
// TrustAwareTemporalGNN on MI455X (gfx1250, wave32, WMMA)
// N=20000 nodes, E=320000 edges, D=64, H=4 (HD=16), L=2 layers, fp32.
// ---------------------------------------------------------------------------
#define NN      20000
#define EE      320000
#define DD      64
#define HH      4
#define HDD     16
#define LL      2
#define SCALE   0.25f          // 1/sqrt(16)
#define EPS_LN  1e-5f

typedef float v2f __attribute__((ext_vector_type(2)));
typedef float v8f __attribute__((ext_vector_type(8)));

// ---------------- wave32 helpers ----------------
__device__ __forceinline__ float wsum32(float v) {
    v += __shfl_xor(v, 16, 32);
    v += __shfl_xor(v,  8, 32);
    v += __shfl_xor(v,  4, 32);
    v += __shfl_xor(v,  2, 32);
    v += __shfl_xor(v,  1, 32);
    return v;
}
__device__ __forceinline__ float wsum8(float v) {   // reduce within groups of 8 lanes
    v += __shfl_xor(v, 4, 32);
    v += __shfl_xor(v, 2, 32);
    v += __shfl_xor(v, 1, 32);
    return v;
}
__device__ __forceinline__ float sigmoidf(float x) { return 1.0f / (1.0f + __expf(-x)); }

__device__ __forceinline__ void atomicMaxF(float* addr, float val) {
    int* iaddr = (int*)addr;
    int old = __float_as_int(*addr);
    while (__int_as_float(old) < val) {
        int assumed = old;
        old = atomicCAS(iaddr, assumed, __float_as_int(val));
        if (old == assumed) break;
    }
}

// ---------------- generic fill ----------------
__global__ void kfill(float* __restrict__ p, long long n, float val) {
    long long i = (long long)blockIdx.x * blockDim.x + threadIdx.x;
    if (i < n) p[i] = val;
}

// ---------------- edge degree count ----------------
__global__ void kcnt(const int* __restrict__ ei, float* __restrict__ cnt) {
    int e = blockIdx.x * blockDim.x + threadIdx.x;
    if (e >= EE) return;
    atomicAdd(&cnt[ei[EE + e]], 1.0f);
}

// ---------------- temporal embedding: sin(t*freq)*exp(-|t|*decay) ----------------
__global__ void ktemb(const float* __restrict__ tdiff, const float* __restrict__ freq,
                      const float* __restrict__ decay, float* __restrict__ temb) {
    long long i = (long long)blockIdx.x * blockDim.x + threadIdx.x;
    if (i >= (long long)EE * DD) return;
    int e = (int)(i >> 6), d = (int)(i & 63);
    float t = tdiff[e];
    temb[i] = __sinf(t * freq[d]) * __expf(-fabsf(t) * decay[0]);
}

// ---------------------------------------------------------------------------
// WMMA GEMM:  C[M x Nc] = A[M x 64] @ W[64 x Nc] + bias   (M % 16 == 0)
// One wave computes one 16x16 tile of C via 16 x V_WMMA_F32_16X16X4_F32.
// ISA layouts (§7.12.2):
//   A 16x4 : lane -> M = lane&15 ; VGPR j -> K = j + 2*(lane>>4)
//   B 4x16 : lane -> N = lane&15 ; VGPR j -> K = j + 2*(lane>>4)
//   C 16x16: lane -> N = lane&15 ; VGPR r -> M = r + 8*(lane>>4)
// Weights are staged in LDS with a paired-K layout sW[p][n][k&1] (p = k>>1)
// so each lane's (kb, kb+1) B-fragment is one ds_load_b64. Each pair-row is
// padded by 32 floats so the half=0 and half=1 lane groups map to disjoint
// 32-bank windows of the 64 LDS banks (conflict-free b64 reads).
// ---------------------------------------------------------------------------
#define RSPAD 32
__global__ __launch_bounds__(128) void
kgemm(const float* __restrict__ A, const float* __restrict__ W,
      const float* __restrict__ bias, float* __restrict__ C, int M, int Nc) {
    __shared__ float sW[32 * (192 * 2 + RSPAD)];   // 32 k-pairs, padded rows
    __shared__ float sB[192];
    const int RS = (Nc << 1) + RSPAD;              // pair-row stride
    int tid = threadIdx.x;
    int tot = 64 * Nc;
    for (int i = tid; i < tot; i += 128) {
        int k = i / Nc, n = i - k * Nc;
        sW[(k >> 1) * RS + (n << 1) + (k & 1)] = W[i];
    }
    for (int i = tid; i < Nc; i += 128) sB[i] = bias[i];
    __syncthreads();

    int wid  = tid >> 5;
    int lane = tid & 31;
    int half = lane >> 4;        // 0 or 1
    int mlo  = lane & 15;
    int row0 = blockIdx.x << 4;
    if (row0 >= M) return;
    const float* Arow = A + (size_t)(row0 + mlo) * 64;

    int NT = Nc >> 4;
    for (int ct = wid; ct < NT; ct += 4) {
        int n0 = ct << 4;
        const float* sWn = sW + ((n0 + mlo) << 1);
        v8f acc = {};
        #pragma unroll
        for (int kk = 0; kk < 16; ++kk) {
            int p  = (kk << 1) + half;               // k-pair index: kb = 2p
            const float2 av = *(const float2*)(Arow + (p << 1));
            const float2 bv = *(const float2*)(sWn + p * RS);
            v2f a; a.x = av.x; a.y = av.y;
            v2f b; b.x = bv.x; b.y = bv.y;
            acc = __builtin_amdgcn_wmma_f32_16x16x4_f32(
                      false, a, false, b, (short)0, acc, false, false);
        }
        float bb = sB[n0 + mlo];
        #pragma unroll
        for (int r = 0; r < 8; ++r) {
            int row = row0 + r + (half << 3);
            C[(size_t)row * Nc + n0 + mlo] = acc[r] + bb;
        }
    }
}

// ---------------- segment-sum of k[src] into ksum[dst] ----------------
__global__ void kksum(const int* __restrict__ ei, const float* __restrict__ kmat,
                      float* __restrict__ ksum) {
    long long i = (long long)blockIdx.x * blockDim.x + threadIdx.x;
    if (i >= (long long)EE * 32) return;
    int e = (int)(i >> 5), t = (int)(i & 31);
    int src = ei[e], dst = ei[EE + e];
    const float2 kv = *(const float2*)(kmat + (size_t)src * 64 + 2 * t);
    float* d = ksum + (size_t)dst * 64 + 2 * t;
    atomicAdd(d,     kv.x);
    atomicAdd(d + 1, kv.y);
}

// ---------------- neigh_mean = ksum / max(cnt,1)  (in place) ----------------
__global__ void knmean(float* __restrict__ ksum, const float* __restrict__ cnt) {
    long long i = (long long)blockIdx.x * blockDim.x + threadIdx.x;
    if (i >= (long long)NN * 64) return;
    ksum[i] /= fmaxf(cnt[i >> 6], 1.0f);
}

// ---------------------------------------------------------------------------
// Edge pass B: attention logits + trust MLP + segment-max. One wave per edge.
// Lane l covers d = 2l, 2l+1 ; head h = l>>3 (lanes 8h..8h+7 own head h).
// ---------------------------------------------------------------------------
__global__ __launch_bounds__(256) void
kedgeB(const int* __restrict__ ei,
       const float* __restrict__ q, const float* __restrict__ kmat,
       const float* __restrict__ th, const float* __restrict__ mem,
       const float* __restrict__ nmean,
       const float* __restrict__ t1W, const float* __restrict__ t1b,
       const float* __restrict__ t2W, const float* __restrict__ t2b,
       float* __restrict__ logits, float* __restrict__ trustb,
       float* __restrict__ segm) {
    int e = blockIdx.x * 8 + (threadIdx.x >> 5);
    if (e >= EE) return;
    int lane = threadIdx.x & 31;
    int src = ei[e], dst = ei[EE + e];
    int d0 = lane << 1;
    int h  = lane >> 3;

    const float2 qv  = *(const float2*)(q     + (size_t)dst * 64 + d0);
    const float2 kv  = *(const float2*)(kmat  + (size_t)src * 64 + d0);
    const float2 tv  = *(const float2*)(th    + (size_t)e   * 64 + d0);
    const float2 mv  = *(const float2*)(mem   + (size_t)dst * 64 + d0);
    const float2 nv  = *(const float2*)(nmean + (size_t)dst * 64 + d0);

    float attn = wsum8(qv.x * (kv.x + tv.x) + qv.y * (kv.y + tv.y)) * SCALE;
    float sim  = wsum32(qv.x * kv.x + qv.y * kv.y) * (1.0f / HH);
    float dx0 = kv.x - mv.x, dx1 = kv.y - mv.y;
    float td   = sqrtf(wsum32(dx0 * dx0 + dx1 * dx1));
    float ex0 = kv.x - nv.x, ex1 = kv.y - nv.y;
    float dev  = sqrtf(wsum32(ex0 * ex0 + ex1 * ex1));

    // trust = sigmoid(relu([td,dev,sim] @ t1_W + t1_b) @ t2_W + t2_b)
    float h0 = fmaxf(0.0f, td * t1W[d0]     + dev * t1W[64 + d0]     + sim * t1W[128 + d0]     + t1b[d0]);
    float h1 = fmaxf(0.0f, td * t1W[d0 + 1] + dev * t1W[64 + d0 + 1] + sim * t1W[128 + d0 + 1] + t1b[d0 + 1]);
    float p0 = wsum32(h0 * t2W[d0 * 4 + 0] + h1 * t2W[(d0 + 1) * 4 + 0]);
    float p1 = wsum32(h0 * t2W[d0 * 4 + 1] + h1 * t2W[(d0 + 1) * 4 + 1]);
    float p2 = wsum32(h0 * t2W[d0 * 4 + 2] + h1 * t2W[(d0 + 1) * 4 + 2]);
    float p3 = wsum32(h0 * t2W[d0 * 4 + 3] + h1 * t2W[(d0 + 1) * 4 + 3]);
    float ph = (h == 0) ? p0 : (h == 1) ? p1 : (h == 2) ? p2 : p3;
    float tr = sigmoidf(ph + t2b[h]);
    float lg = attn * tr;

    if ((lane & 7) == 0) {
        logits[(size_t)e * 4 + h] = lg;
        trustb[(size_t)e * 4 + h] = tr;
        atomicMaxF(&segm[(size_t)dst * 4 + h], lg);
    }
}

// ---------------------------------------------------------------------------
// Edge pass C: exp(logit-max), GRU cell, scatter numerator/denominator.
// ---------------------------------------------------------------------------
__global__ __launch_bounds__(256) void
kedgeC(const int* __restrict__ ei,
       const float* __restrict__ logits, const float* __restrict__ trustb,
       const float* __restrict__ segm,
       const float* __restrict__ gi, const float* __restrict__ gh,
       const float* __restrict__ mem,
       float* __restrict__ num, float* __restrict__ denom) {
    int e = blockIdx.x * 8 + (threadIdx.x >> 5);
    if (e >= EE) return;
    int lane = threadIdx.x & 31;
    int src = ei[e], dst = ei[EE + e];
    int d0 = lane << 1;
    int h  = lane >> 3;

    float eh = __expf(logits[(size_t)e * 4 + h] - segm[(size_t)dst * 4 + h]);
    float tr = trustb[(size_t)e * 4 + h];

    const float* gis = gi + (size_t)src * 192;
    const float* ghd = gh + (size_t)dst * 192;
    const float2 ir = *(const float2*)(gis + d0);
    const float2 iz = *(const float2*)(gis + 64 + d0);
    const float2 in = *(const float2*)(gis + 128 + d0);
    const float2 hr = *(const float2*)(ghd + d0);
    const float2 hz = *(const float2*)(ghd + 64 + d0);
    const float2 hn = *(const float2*)(ghd + 128 + d0);
    const float2 mj = *(const float2*)(mem + (size_t)dst * 64 + d0);

    float r0 = sigmoidf(ir.x + hr.x), r1 = sigmoidf(ir.y + hr.y);
    float z0 = sigmoidf(iz.x + hz.x), z1 = sigmoidf(iz.y + hz.y);
    float n0 = tanhf(in.x + r0 * hn.x), n1 = tanhf(in.y + r1 * hn.y);
    float hc0 = (1.0f - z0) * n0 + z0 * mj.x;
    float hc1 = (1.0f - z1) * n1 + z1 * mj.y;

    float w = eh * tr;
    float* np = num + (size_t)dst * 64 + d0;
    atomicAdd(np,     hc0 * w);
    atomicAdd(np + 1, hc1 * w);
    if ((lane & 7) == 0) atomicAdd(&denom[(size_t)dst * 4 + h], eh);
}

// ---------------------------------------------------------------------------
// Node pass: out = num/denom ; LayerNorm(h + out). One wave per node.
// ---------------------------------------------------------------------------
__global__ __launch_bounds__(256) void
knodeOut(const float* __restrict__ num, const float* __restrict__ denom,
         const float* __restrict__ hin, const float* __restrict__ g,
         const float* __restrict__ b, float* __restrict__ hout) {
    int node = blockIdx.x * 8 + (threadIdx.x >> 5);
    if (node >= NN) return;
    int lane = threadIdx.x & 31;
    int d0 = lane << 1;
    int h  = lane >> 3;

    const float2 nu = *(const float2*)(num + (size_t)node * 64 + d0);
    float den = denom[(size_t)node * 4 + h] + 1e-16f;
    const float2 hv = *(const float2*)(hin + (size_t)node * 64 + d0);
    float x0 = hv.x + nu.x / den;
    float x1 = hv.y + nu.y / den;

    float mu = wsum32(x0 + x1) * (1.0f / 64.0f);
    float q0 = x0 - mu, q1 = x1 - mu;
    float var = wsum32(q0 * q0 + q1 * q1) * (1.0f / 64.0f);
    float rs = rsqrtf(var + EPS_LN);

    float2 o;
    o.x = q0 * rs * g[d0]     + b[d0];
    o.y = q1 * rs * g[d0 + 1] + b[d0 + 1];
    *(float2*)(hout + (size_t)node * 64 + d0) = o;
}

// ---------------------------------------------------------------------------
// Host launcher
// ---------------------------------------------------------------------------
extern "C" void kernel_launch(void* const* d_in, const int* in_sizes, int n_in,
                              void* d_out, int out_size, void* d_ws, size_t ws_size,
                              hipStream_t stream) {
    const float* x      = (const float*)d_in[0];
    const int*   ei     = (const int*)  d_in[1];
    const float* tdiff  = (const float*)d_in[2];
    const float* mem    = (const float*)d_in[3];
    const float* projW  = (const float*)d_in[4];
    const float* projB  = (const float*)d_in[5];
    const float* freq   = (const float*)d_in[6];
    const float* decay  = (const float*)d_in[7];
    const float* lqW    = (const float*)d_in[8];
    const float* lqB    = (const float*)d_in[9];
    const float* lkW    = (const float*)d_in[10];
    const float* lkB    = (const float*)d_in[11];
    const float* lvW    = (const float*)d_in[12];
    const float* lvB    = (const float*)d_in[13];
    const float* ltW    = (const float*)d_in[14];
    const float* ltB    = (const float*)d_in[15];
    const float* t1W    = (const float*)d_in[16];
    const float* t1B    = (const float*)d_in[17];
    const float* t2W    = (const float*)d_in[18];
    const float* t2B    = (const float*)d_in[19];
    const float* gWih   = (const float*)d_in[20];
    const float* gWhh   = (const float*)d_in[21];
    const float* gBih   = (const float*)d_in[22];
    const float* gBhh   = (const float*)d_in[23];
    const float* lnG    = (const float*)d_in[24];
    const float* lnB    = (const float*)d_in[25];

    // ---- workspace carve-up (floats) ----
    float* w = (float*)d_ws;
    float* hbuf  = w;                 w += (size_t)NN * 64;
    float* temb  = w;                 w += (size_t)EE * 64;
    float* thb   = w;                 w += (size_t)EE * 64;
    float* qb    = w;                 w += (size_t)NN * 64;
    float* kb    = w;                 w += (size_t)NN * 64;
    float* vb    = w;                 w += (size_t)NN * 64;
    float* gib   = w;                 w += (size_t)NN * 192;
    float* ghb   = w;                 w += (size_t)NN * 192;
    float* ksum  = w;                 w += (size_t)NN * 64;
    float* cnt   = w;                 w += (size_t)NN;
    float* logb  = w;                 w += (size_t)EE * 4;
    float* trub  = w;                 w += (size_t)EE * 4;
    float* segm  = w;                 w += (size_t)NN * 4;
    float* numb  = w;                 w += (size_t)NN * 64;
    float* denb  = w;                 w += (size_t)NN * 4;

    const int TB = 256;
    auto blocks = [](long long n, int t) { return (int)((n + t - 1) / t); };

    // degree counts (layer-invariant)
    kfill<<<blocks(NN, TB), TB, 0, stream>>>(cnt, NN, 0.0f);
    kcnt <<<blocks(EE, TB), TB, 0, stream>>>(ei, cnt);

    // temporal embedding
    ktemb<<<blocks((long long)EE * 64, TB), TB, 0, stream>>>(tdiff, freq, decay, temb);

    // input projection: h = x @ proj_W + proj_b
    kgemm<<<NN / 16, 128, 0, stream>>>(x, projW, projB, hbuf, NN, 64);

    for (int l = 0; l < LL; ++l) {
        const float* WQ = lqW + (size_t)l * 64 * 64;
        const float* WK = lkW + (size_t)l * 64 * 64;
        const float* WV = lvW + (size_t)l * 64 * 64;
        const float* WT = ltW + (size_t)l * 64 * 64;
        const float* Wi = gWih + (size_t)l * 64 * 192;
        const float* Wh = gWhh + (size_t)l * 64 * 192;

        // node projections (WMMA)
        kgemm<<<NN / 16, 128, 0, stream>>>(hbuf, WQ, lqB + l * 64, qb, NN, 64);
        kgemm<<<NN / 16, 128, 0, stream>>>(hbuf, WK, lkB + l * 64, kb, NN, 64);
        kgemm<<<NN / 16, 128, 0, stream>>>(hbuf, WV, lvB + l * 64, vb, NN, 64);
        kgemm<<<NN / 16, 128, 0, stream>>>(mem,  Wh, gBhh + l * 192, ghb, NN, 192);
        kgemm<<<NN / 16, 128, 0, stream>>>(vb,   Wi, gBih + l * 192, gib, NN, 192);
        // edge temporal projection (WMMA, E rows)
        kgemm<<<EE / 16, 128, 0, stream>>>(temb, WT, ltB + l * 64, thb, EE, 64);

        // neighborhood mean of k over dst
        kfill<<<blocks((long long)NN * 64, TB), TB, 0, stream>>>(ksum, (long long)NN * 64, 0.0f);
        kksum<<<blocks((long long)EE * 32, TB), TB, 0, stream>>>(ei, kb, ksum);
        knmean<<<blocks((long long)NN * 64, TB), TB, 0, stream>>>(ksum, cnt);

        // logits + trust + segment max
        kfill<<<blocks((long long)NN * 4, TB), TB, 0, stream>>>(segm, (long long)NN * 4, -3.0e38f);
        kedgeB<<<EE / 8, 256, 0, stream>>>(ei, qb, kb, thb, mem, ksum,
                                           t1W + (size_t)l * 192, t1B + l * 64,
                                           t2W + (size_t)l * 256, t2B + l * 4,
                                           logb, trub, segm);

        // softmax-weighted GRU messages
        kfill<<<blocks((long long)NN * 64, TB), TB, 0, stream>>>(numb, (long long)NN * 64, 0.0f);
        kfill<<<blocks((long long)NN * 4, TB), TB, 0, stream>>>(denb, (long long)NN * 4, 0.0f);
        kedgeC<<<EE / 8, 256, 0, stream>>>(ei, logb, trub, segm, gib, ghb, mem, numb, denb);

        // normalize + residual + LayerNorm
        float* outp = (l == LL - 1) ? (float*)d_out : hbuf;
        knodeOut<<<NN / 8, 256, 0, stream>>>(numb, denb, hbuf, lnG + l * 64, lnB + l * 64, outp);
    }
}